// ShapleyFusion_62534723829925
// MI455X (gfx1250) — compile-verified
//
#include <hip/hip_runtime.h>

typedef __attribute__((ext_vector_type(2))) float v2f;
typedef __attribute__((ext_vector_type(8))) float v8f;

#define LL 5
#define CC 256
#define NN (128 * 384)          // H*W = 49152
#define WAVES_PER_BLOCK 8
#define PIX_PER_WAVE 32

__global__ __launch_bounds__(256)
void shapley_fused_kernel(const float* __restrict__ feats,
                          const float* __restrict__ w,
                          const float* __restrict__ bias,
                          float* __restrict__ out)
{
    const int lane = threadIdx.x & 31;
    const int wib  = threadIdx.x >> 5;                    // wave in block
    const int waveId = blockIdx.x * WAVES_PER_BLOCK + wib;
    const int n0 = waveId * PIX_PER_WAVE;                 // first pixel of this wave's tile

    const int m15 = lane & 15;   // pixel row within 16-wide WMMA tile
    const int hi  = lane >> 4;   // 0: lanes 0-15 (K=khi,khi+1), 1: lanes 16-31 (K=khi+2,khi+3)
    const int khi = hi * 2;

    __shared__ float zbuf[WAVES_PER_BLOCK][LL][PIX_PER_WAVE];

    // ---- Phase 1: z[l, pixel] via V_WMMA_F32_16X16X4_F32 --------------------
    // D = A(16x4) x B(4x16) + C, accumulated over 64 K-chunks of 4 channels.
    // B columns all equal w[k] -> every D column equals the z vector.
    v8f acc[LL][2];
    #pragma unroll
    for (int l = 0; l < LL; ++l) {
        #pragma unroll
        for (int t = 0; t < 2; ++t) {
            acc[l][t] = (v8f)0.0f;
        }
    }

    for (int c0 = 0; c0 < CC; c0 += 4) {
        // B-matrix: VGPR0 = w[c0+khi] (lanes 0-15) / w[c0+khi+2] (lanes 16-31),
        //           VGPR1 = w[c0+khi+1] / w[c0+khi+3]  -> contiguous b64 per lane.
        v2f bv;
        bv.x = w[c0 + khi];
        bv.y = w[c0 + khi + 1];

        #pragma unroll
        for (int l = 0; l < LL; ++l) {
            const float* fp = feats + (size_t)(l * CC + c0 + khi) * NN + n0 + m15;
            // A-matrix, pixel tile 0 (pixels n0+0..15): VGPR0 = K=khi row, VGPR1 = K=khi+1 row
            v2f a0; a0.x = fp[0];  a0.y = fp[NN];
            // A-matrix, pixel tile 1 (pixels n0+16..31)
            v2f a1; a1.x = fp[16]; a1.y = fp[NN + 16];

            acc[l][0] = __builtin_amdgcn_wmma_f32_16x16x4_f32(
                false, a0, false, bv, (short)0, acc[l][0], false, false);
            acc[l][1] = __builtin_amdgcn_wmma_f32_16x16x4_f32(
                false, a1, false, bv, (short)0, acc[l][1], false, false);
        }
    }

    // ---- Redistribute z through LDS ----------------------------------------
    // D layout: VGPR v, lanes 0-15 -> row M=v, lanes 16-31 -> row M=v+8.
    // All 16 lanes of a half-wave hold identical values; let lanes 0 and 16 write.
    if (m15 == 0) {
        #pragma unroll
        for (int l = 0; l < LL; ++l) {
            #pragma unroll
            for (int t = 0; t < 2; ++t) {
                #pragma unroll
                for (int i = 0; i < 8; ++i) {
                    zbuf[wib][l][t * 16 + hi * 8 + i] = acc[l][t][i];
                }
            }
        }
    }
    __syncthreads();

    // ---- Phase 2: per-pixel Shapley + softmax (lane == pixel) --------------
    float z[LL];
    #pragma unroll
    for (int l = 0; l < LL; ++l) z[l] = zbuf[wib][l][lane];
    const float bb = bias[0];

    // wgt(card) = card! * (4-card)! / 5!
    const float wtab1 = 0.05f;          // card 1
    const float wtab2 = 1.0f / 30.0f;   // card 2
    const float wtab3 = 0.05f;          // card 3
    const float wtab4 = 0.2f;           // card 4

    float shap[LL];
    #pragma unroll
    for (int i = 0; i < LL; ++i) {
        float accs = 0.0f;
        float sumw = 0.0f;
        #pragma unroll
        for (int m = 1; m < 16; ++m) {
            float s = z[i];
            int card = 0;
            #pragma unroll
            for (int k = 0; k < 4; ++k) {
                if (m & (1 << k)) {
                    const int j = k + (k >= i ? 1 : 0);   // k-th "other" agent
                    s += z[j];
                    ++card;
                }
            }
            const float wv = (card == 1) ? wtab1 :
                             (card == 2) ? wtab2 :
                             (card == 3) ? wtab3 : wtab4;
            sumw += wv;
            accs = fmaf(wv, fmaxf(s + bb, 0.0f), accs);
        }
        const float sv = fmaxf(z[i] + bb, 0.0f);
        shap[i] = accs - sumw * sv + sv * 0.2f;           // + sv / L, L = 5
    }

    // softmax over agents
    float mx = shap[0];
    #pragma unroll
    for (int i = 1; i < LL; ++i) mx = fmaxf(mx, shap[i]);
    float e[LL];
    float es = 0.0f;
    #pragma unroll
    for (int i = 0; i < LL; ++i) { e[i] = __expf(shap[i] - mx); es += e[i]; }
    const float inv = 1.0f / es;
    float ws[LL];
    #pragma unroll
    for (int i = 0; i < LL; ++i) ws[i] = e[i] * inv;

    // ---- Phase 3: out[c, n] = sum_l feats[l,c,n] * ws[l] -------------------
    // Re-read is L2-resident (same addresses this wave just streamed).
    const int n = n0 + lane;
    for (int c = 0; c < CC; ++c) {
        float s = 0.0f;
        #pragma unroll
        for (int l = 0; l < LL; ++l) {
            s = fmaf(feats[(size_t)(l * CC + c) * NN + n], ws[l], s);
        }
        out[(size_t)c * NN + n] = s;
    }
}

extern "C" void kernel_launch(void* const* d_in, const int* in_sizes, int n_in,
                              void* d_out, int out_size, void* d_ws, size_t ws_size,
                              hipStream_t stream) {
    const float* feats = (const float*)d_in[0];   // (L, C, H, W) fp32
    const float* w     = (const float*)d_in[1];   // (C, 1) fp32
    const float* bias  = (const float*)d_in[2];   // (1,) fp32
    float* out = (float*)d_out;                   // (C, H, W) fp32

    const int waves  = NN / PIX_PER_WAVE;         // 1536
    const int blocks = waves / WAVES_PER_BLOCK;   // 192
    shapley_fused_kernel<<<blocks, 256, 0, stream>>>(feats, w, bias, out);
}